// SSIM_9354438771010
// MI455X (gfx1250) — compile-verified
//
#include <hip/hip_runtime.h>
#include <math.h>

typedef __attribute__((ext_vector_type(2))) float v2f;
typedef __attribute__((ext_vector_type(8))) float v8f;

#define TILE   16
#define HALO   5
#define IN_R   26          // TILE + 2*HALO
#define LDS_R  32          // padded rows (zero-filled beyond 26)
#define LDS_C  28          // padded cols (zero-filled beyond 26), K padded to 28
#define KSTEPS 7           // 28 / 4
#define NWAVE  4
#define IMG_H  512
#define IMG_W  512
#define TX     (IMG_W / TILE)   // 32
#define TY     (IMG_H / TILE)   // 32
#define NBATCH 64
#define NTILES (NBATCH * TX * TY)      // 65536
#define NBLOCKS (NTILES / NWAVE)       // 16384

__device__ __forceinline__ v8f wmma4(v2f a, v2f b, v8f c) {
    // D = A(16x4,f32) * B(4x16,f32) + C(16x16,f32)
    return __builtin_amdgcn_wmma_f32_16x16x4_f32(
        /*neg_a=*/false, a, /*neg_b=*/false, b,
        /*c_mod=*/(short)0, c, /*reuse_a=*/false, /*reuse_b=*/false);
}

__global__ __launch_bounds__(NWAVE * 32)
void ssim_wmma_kernel(const float* __restrict__ x,
                      const float* __restrict__ y,
                      const float* __restrict__ win,
                      float* __restrict__ partial)
{
    __shared__ float Xs[NWAVE][LDS_R][LDS_C];
    __shared__ float Ys[NWAVE][LDS_R][LDS_C];
    __shared__ float Hc[NWAVE][LDS_R][TILE];
    __shared__ float gws[16];
    __shared__ float wsum[NWAVE];

    const int wv   = threadIdx.x >> 5;
    const int lane = threadIdx.x & 31;
    const int hl   = lane >> 4;   // lane half: 0 -> lanes 0-15, 1 -> lanes 16-31
    const int l15  = lane & 15;

    // Recover the 1-D gaussian from the actual window input: w2d[i][i] = g[i]^2
    if (threadIdx.x < 11)  gws[threadIdx.x] = sqrtf(win[threadIdx.x * 12]);
    if (threadIdx.x >= 11 && threadIdx.x < 16) gws[threadIdx.x] = 0.0f;

    // One 16x16 output tile per wave
    const int tileId = blockIdx.x * NWAVE + wv;
    const int b  = tileId / (TX * TY);
    const int t2 = tileId - b * (TX * TY);
    const int ty = t2 >> 5;   // / TX  (TX == 32)
    const int tx = t2 & 31;   // % TX
    const int r0 = ty * TILE, c0 = tx * TILE;
    const float* xb = x + (size_t)b * (IMG_H * IMG_W);
    const float* yb = y + (size_t)b * (IMG_H * IMG_W);

    // Load zero-padded halo tiles into LDS (26x26 valid inside 32x28)
    for (int idx = lane; idx < LDS_R * LDS_C; idx += 32) {
        const int i  = idx / LDS_C;
        const int j  = idx - i * LDS_C;
        const int gr = r0 - HALO + i;
        const int gc = c0 - HALO + j;
        const bool ok = (i < IN_R) && (j < IN_R) &&
                        (gr >= 0) && (gr < IMG_H) && (gc >= 0) && (gc < IMG_W);
        float xv = 0.0f, yv = 0.0f;
        if (ok) {
            xv = xb[gr * IMG_W + gc];
            yv = yb[gr * IMG_W + gc];
        }
        Xs[wv][i][j] = xv;
        Ys[wv][i][j] = yv;
    }
    __syncthreads();

    // Per-lane banded-weight fragments.
    // Stage-1 B operand: Wh[j,n] = g[j-n]; rows k0+hl and k0+2+hl, col = l15.
    // Stage-2 A operand: Wv[m,h] = g[h-m]; m = l15, cols kb, kb+1.
    v2f whB[KSTEPS], wvA[KSTEPS];
#pragma unroll
    for (int ks = 0; ks < KSTEPS; ++ks) {
        const int k0 = 4 * ks;
        const int jb = k0 + hl;
        whB[ks].x = ((unsigned)(jb - l15)     <= 10u) ? gws[jb - l15]     : 0.0f;
        whB[ks].y = ((unsigned)(jb + 2 - l15) <= 10u) ? gws[jb + 2 - l15] : 0.0f;
        const int kb = k0 + 2 * hl;
        wvA[ks].x = ((unsigned)(kb - l15)     <= 10u) ? gws[kb - l15]     : 0.0f;
        wvA[ks].y = ((unsigned)(kb + 1 - l15) <= 10u) ? gws[kb + 1 - l15] : 0.0f;
    }

    // Channels: 0:x  1:y  2:x*x  3:y*y  4:x*y
    v8f outAcc[5];
#pragma unroll
    for (int ch = 0; ch < 5; ++ch) {
        // ---- Stage 1: horizontal conv as GEMM  Hc(26x16) = T(26x28) * Wh(28x16)
#pragma unroll
        for (int mb = 0; mb < 2; ++mb) {
            const int row = mb * TILE + l15;
            v8f acc = {};
#pragma unroll
            for (int ks = 0; ks < KSTEPS; ++ks) {
                const int kb = 4 * ks + 2 * hl;
                const float x0 = Xs[wv][row][kb], x1 = Xs[wv][row][kb + 1];
                const float y0 = Ys[wv][row][kb], y1 = Ys[wv][row][kb + 1];
                v2f a;
                if      (ch == 0) { a.x = x0;      a.y = x1;      }
                else if (ch == 1) { a.x = y0;      a.y = y1;      }
                else if (ch == 2) { a.x = x0 * x0; a.y = x1 * x1; }
                else if (ch == 3) { a.x = y0 * y0; a.y = y1 * y1; }
                else              { a.x = x0 * y0; a.y = x1 * y1; }
                acc = wmma4(a, whB[ks], acc);
            }
            // D layout: VGPR i -> M = mb*16 + i (+8 for upper lane half), N = l15
#pragma unroll
            for (int i = 0; i < 8; ++i)
                Hc[wv][mb * TILE + hl * 8 + i][l15] = acc[i];
        }
        __syncthreads();

        // ---- Stage 2: vertical conv as GEMM  Out(16x16) = Wv(16x28) * Hc(28x16)
        v8f o = {};
#pragma unroll
        for (int ks = 0; ks < KSTEPS; ++ks) {
            const int k0 = 4 * ks;
            v2f bf;
            bf.x = Hc[wv][k0 + hl][l15];
            bf.y = Hc[wv][k0 + 2 + hl][l15];
            o = wmma4(wvA[ks], bf, o);
        }
        outAcc[ch] = o;
        __syncthreads();   // Hc reused by next channel
    }

    // ---- SSIM map + reduction
    const float C1v = 0.01f * 0.01f;
    const float C2v = 0.03f * 0.03f;
    float s = 0.0f;
#pragma unroll
    for (int i = 0; i < 8; ++i) {
        const float mx  = outAcc[0][i], my  = outAcc[1][i];
        const float sxx = outAcc[2][i], syy = outAcc[3][i], sxy = outAcc[4][i];
        const float mx2 = mx * mx, my2 = my * my, mxy = mx * my;
        const float num = (2.0f * mxy + C1v) * (2.0f * (sxy - mxy) + C2v);
        const float den = (mx2 + my2 + C1v) * ((sxx - mx2) + (syy - my2) + C2v);
        s += num / den;
    }
#pragma unroll
    for (int off = 16; off > 0; off >>= 1)
        s += __shfl_xor(s, off, 32);
    if (lane == 0) wsum[wv] = s;
    __syncthreads();
    if (threadIdx.x == 0)
        partial[blockIdx.x] = wsum[0] + wsum[1] + wsum[2] + wsum[3];
}

__global__ __launch_bounds__(256)
void ssim_reduce_kernel(const float* __restrict__ partial, int n,
                        float* __restrict__ out)
{
    __shared__ float sm[256];
    float s = 0.0f;
    for (int i = threadIdx.x; i < n; i += 256) s += partial[i];
    sm[threadIdx.x] = s;
    __syncthreads();
    for (int st = 128; st > 0; st >>= 1) {
        if (threadIdx.x < st) sm[threadIdx.x] += sm[threadIdx.x + st];
        __syncthreads();
    }
    if (threadIdx.x == 0)
        out[0] = sm[0] * (1.0f / (float)(NBATCH * IMG_H * IMG_W));
}

extern "C" void kernel_launch(void* const* d_in, const int* in_sizes, int n_in,
                              void* d_out, int out_size, void* d_ws, size_t ws_size,
                              hipStream_t stream) {
    const float* x   = (const float*)d_in[0];
    const float* y   = (const float*)d_in[1];
    const float* win = (const float*)d_in[2];
    float* out       = (float*)d_out;
    float* partial   = (float*)d_ws;   // NBLOCKS floats = 64 KB

    ssim_wmma_kernel<<<NBLOCKS, NWAVE * 32, 0, stream>>>(x, y, win, partial);
    ssim_reduce_kernel<<<1, 256, 0, stream>>>(partial, NBLOCKS, out);
}